// ConstraintAwareLoss_80676665688367
// MI455X (gfx1250) — compile-verified
//
#include <hip/hip_runtime.h>

// ---------------------------------------------------------------------------
// ConstraintAwareLoss for MI455X (gfx1250, wave32).
// Single-pass streaming reduction (~136 MB -> ~6 us @ 23.3 TB/s HBM).
// Cross-lane reductions use V_WMMA_F32_16X16X4_F32 (A=data, B=ones), which is
// layout-invariant: sum of all A slots == sum over D's 8 VGPRs + shfl_xor(16).
// ---------------------------------------------------------------------------

#define NBLK     1024
#define THREADS  256
#define S_LEN    16384           // sequence length (fixed by reference setup)
#define GROW     (S_LEN / 4)     // groups per row = 4096

typedef __attribute__((ext_vector_type(2))) float v2f;
typedef __attribute__((ext_vector_type(8))) float v8f;

#define PACK_SPAN(lo, hi) ((int)((((hi) & 0xFFFF) << 16) | ((lo) & 0xFFFF)))

// MIN_SPAN (low 16, signed) / MAX_SPAN (high 16) packed, row-major [f1][f2].
__constant__ int c_span[25] = {
  PACK_SPAN(  0, 0), PACK_SPAN( -5,10), PACK_SPAN( -4,12), PACK_SPAN( -3,14), PACK_SPAN( -1,16),
  PACK_SPAN(-10, 5), PACK_SPAN(  0, 0), PACK_SPAN(  1, 8), PACK_SPAN(  2,10), PACK_SPAN(  3,12),
  PACK_SPAN(-12, 4), PACK_SPAN( -1, 1), PACK_SPAN(  0, 0), PACK_SPAN(  1, 8), PACK_SPAN(  2,10),
  PACK_SPAN(-14, 3), PACK_SPAN( -2, 1), PACK_SPAN( -1, 1), PACK_SPAN(  0, 0), PACK_SPAN(  1, 8),
  PACK_SPAN(-16, 1), PACK_SPAN( -3, 1), PACK_SPAN( -2, 1), PACK_SPAN( -1, 1), PACK_SPAN(  0, 0)
};

// Exact 32-lane sum via the matrix pipe.
// A (16x4 f32, 2 VGPRs/lane): slot0 = x, slot1 = 0.  B = all ones.
// D[m][n] = rowsum(m); D's VGPR j holds rowsum(j) (lanes 0-15) / rowsum(j+8)
// (lanes 16-31), so sum(d[0..7]) + shfl_xor(16) = sum over every A slot = sum(x).
// Correct for ANY (lane,vgpr)->(m,k) bijection, so robust to layout details.
// Requires EXEC all-ones (callers are fully converged).
__device__ __forceinline__ float wave_sum32(float x) {
  v2f a; a.x = x;    a.y = 0.0f;
  v2f b; b.x = 1.0f; b.y = 1.0f;
  v8f c = {};
  v8f d = __builtin_amdgcn_wmma_f32_16x16x4_f32(
      /*neg_a=*/false, a, /*neg_b=*/false, b,
      /*c_mod=*/(short)0, c, /*reuse_a=*/false, /*reuse_b=*/false);
  float r = ((d[0] + d[1]) + (d[2] + d[3])) + ((d[4] + d[5]) + (d[6] + d[7]));
  r += __shfl_xor(r, 16);
  return r;
}

__device__ __forceinline__ int argmax5(float a0, float a1, float a2, float a3, float a4) {
  int p = 0; float b = a0;               // strict '>' => first max wins (jnp.argmax)
  if (a1 > b) { b = a1; p = 1; }
  if (a2 > b) { b = a2; p = 2; }
  if (a3 > b) { b = a3; p = 3; }
  if (a4 > b) { p = 4; }
  return p;
}

__global__ __launch_bounds__(THREADS)
void caw_main_kernel(const float* __restrict__ logits,
                     const int* __restrict__ labels,
                     const int* __restrict__ pitches,
                     const int* __restrict__ hands,
                     const unsigned char* __restrict__ mask,
                     float* __restrict__ ws, int NG) {
  const int tid      = blockIdx.x * THREADS + threadIdx.x;
  const int nthreads = gridDim.x * THREADS;
  const int lane     = threadIdx.x & 31;
  const int wid      = threadIdx.x >> 5;

  float acc_ce = 0.0f, acc_m = 0.0f, acc_v = 0.0f, acc_t = 0.0f;

  // Each group g = 4 consecutive positions. Waves cover 32 consecutive groups
  // (128 positions), never straddling a row (16384 % 128 == 0). NG % 32 == 0,
  // so waves drop out of the loop whole -> shuffles below are well-defined.
  for (int g = tid; g < NG; g += nthreads) {
    const size_t i = (size_t)g * 4;

    float4 v4[5];                               // 20 logits, 16B-aligned (g*80)
    const float4* lp = (const float4*)(logits + i * 5);
#pragma unroll
    for (int k = 0; k < 5; ++k) v4[k] = lp[k];
    const float* v = (const float*)v4;

    const int4   lab = ((const int4*)labels)[g];
    const int4   pit = ((const int4*)pitches)[g];
    const int4   hnd = ((const int4*)hands)[g];
    const uchar4 msk = ((const uchar4*)mask)[g];

    const int labv[4] = { lab.x, lab.y, lab.z, lab.w };
    const int hv[4]   = { hnd.x, hnd.y, hnd.z, hnd.w };
    int pv[5]; pv[0] = pit.x; pv[1] = pit.y; pv[2] = pit.z; pv[3] = pit.w;
    int mv[5]; mv[0] = msk.x; mv[1] = msk.y; mv[2] = msk.z; mv[3] = msk.w;

    int pr[5];
#pragma unroll
    for (int j = 0; j < 4; ++j) {
      const float a0 = v[j*5+0], a1 = v[j*5+1], a2 = v[j*5+2],
                  a3 = v[j*5+3], a4 = v[j*5+4];
      // masked cross-entropy: lse - logit[label]
      const float mx  = fmaxf(fmaxf(fmaxf(a0, a1), fmaxf(a2, a3)), a4);
      const float se  = __expf(a0-mx) + __expf(a1-mx) + __expf(a2-mx)
                      + __expf(a3-mx) + __expf(a4-mx);
      const float lse = mx + __logf(se);
      const int   lb  = labv[j];
      const float sel = (lb == 0) ? a0 : (lb == 1) ? a1 : (lb == 2) ? a2
                      : (lb == 3) ? a3 : a4;
      const float mm  = (float)mv[j];
      acc_ce += (lse - sel) * mm;
      acc_m  += mm;
      pr[j] = argmax5(a0, a1, a2, a3, a4);
    }

    // Next group's first pitch/mask/prediction (lane L+1 has it; lane 31
    // re-derives it, and is also the only lane that can be at a row end).
    int pn  = __shfl_down(pv[0], 1);
    int mn  = __shfl_down(mv[0], 1);
    int prn = __shfl_down(pr[0], 1);
    const bool rowEnd = ((g & (GROW - 1)) == (GROW - 1));
    if (lane == 31 && !rowEnd) {
      const size_t ni = i + 4;
      pn = pitches[ni];
      mn = mask[ni];
      const float* q = logits + ni * 5;
      prn = argmax5(q[0], q[1], q[2], q[3], q[4]);
    }
    pv[4] = pn; mv[4] = mn; pr[4] = prn;

#pragma unroll
    for (int j = 0; j < 4; ++j) {
      const bool active  = (j < 3) | (!rowEnd);
      const int interval = pv[j+1] - pv[j];
      const int eff      = (hv[j] == 1) ? interval : -interval;
      const int e        = c_span[pr[j] * 5 + pr[j+1]];
      const int lo       = (int)(short)(e & 0xFFFF);
      const int hi       = e >> 16;
      const bool valid   = (eff >= lo) & (eff <= hi);
      const bool tm      = active & (mv[j] != 0) & (mv[j+1] != 0);
      acc_t += tm ? 1.0f : 0.0f;
      acc_v += (tm & !valid) ? 1.0f : 0.0f;
    }
  }

  // Wave reduction on the matrix pipe (EXEC all-ones here), then block sum.
  const float w0 = wave_sum32(acc_ce);
  const float w1 = wave_sum32(acc_m);
  const float w2 = wave_sum32(acc_v);
  const float w3 = wave_sum32(acc_t);

  __shared__ float sred[THREADS / 32][4];
  if (lane == 0) {
    sred[wid][0] = w0; sred[wid][1] = w1; sred[wid][2] = w2; sred[wid][3] = w3;
  }
  __syncthreads();
  if (threadIdx.x == 0) {
    float t0 = 0.f, t1 = 0.f, t2 = 0.f, t3 = 0.f;
#pragma unroll
    for (int w = 0; w < THREADS / 32; ++w) {
      t0 += sred[w][0]; t1 += sred[w][1]; t2 += sred[w][2]; t3 += sred[w][3];
    }
    ws[0 * NBLK + blockIdx.x] = t0;
    ws[1 * NBLK + blockIdx.x] = t1;
    ws[2 * NBLK + blockIdx.x] = t2;
    ws[3 * NBLK + blockIdx.x] = t3;
  }
}

__global__ __launch_bounds__(THREADS)
void caw_finalize_kernel(const float* __restrict__ ws, float* __restrict__ out) {
  const int t    = threadIdx.x;
  const int lane = t & 31;
  const int wid  = t >> 5;

  float part[4];
#pragma unroll
  for (int q = 0; q < 4; ++q) {
    float s = 0.0f;
#pragma unroll
    for (int k = 0; k < NBLK / THREADS; ++k) s += ws[q * NBLK + t + THREADS * k];
    part[q] = wave_sum32(s);
  }

  __shared__ float sred[THREADS / 32][4];
  if (lane == 0) {
#pragma unroll
    for (int q = 0; q < 4; ++q) sred[wid][q] = part[q];
  }
  __syncthreads();
  if (t == 0) {
    float tot[4] = {0.f, 0.f, 0.f, 0.f};
#pragma unroll
    for (int w = 0; w < THREADS / 32; ++w)
#pragma unroll
      for (int q = 0; q < 4; ++q) tot[q] += sred[w][q];

    const float ce_loss = tot[0] / tot[1];
    const float c_loss  = (tot[3] > 0.0f) ? (tot[2] / fmaxf(tot[3], 1.0f)) : 0.0f;
    out[0] = ce_loss + 0.1f * c_loss;
  }
}

extern "C" void kernel_launch(void* const* d_in, const int* in_sizes, int n_in,
                              void* d_out, int out_size, void* d_ws, size_t ws_size,
                              hipStream_t stream) {
  const float*         logits  = (const float*)d_in[0];
  const int*           labels  = (const int*)d_in[1];
  const int*           pitches = (const int*)d_in[2];
  const int*           hands   = (const int*)d_in[3];
  const unsigned char* mask    = (const unsigned char*)d_in[4];  // jnp bool = 1 byte
  float* out = (float*)d_out;
  float* ws  = (float*)d_ws;   // 4 * NBLK floats = 16 KB, fully rewritten each call

  const int N  = in_sizes[1];  // B * S = 4,194,304
  const int NG = N / 4;        // groups of 4 positions

  caw_main_kernel<<<NBLK, THREADS, 0, stream>>>(logits, labels, pitches, hands,
                                                mask, ws, NG);
  caw_finalize_kernel<<<1, THREADS, 0, stream>>>(ws, out);
}